// DiffDMC_27960237097003
// MI455X (gfx1250) — compile-verified
//
#include <hip/hip_runtime.h>
#include <stdint.h>

#define TB 256
#define EPSF 1e-12f

// ---------------- small float3 helpers ----------------
struct F3 { float x, y, z; };
__device__ __forceinline__ F3 f3sub(F3 a, F3 b) { return {a.x - b.x, a.y - b.y, a.z - b.z}; }
__device__ __forceinline__ float f3dot(F3 a, F3 b) { return a.x * b.x + a.y * b.y + a.z * b.z; }
__device__ __forceinline__ F3 f3cross(F3 a, F3 b) {
    return {a.y * b.z - a.z * b.y, a.z * b.x - a.x * b.z, a.x * b.y - a.y * b.x};
}
__device__ __forceinline__ float rlen(F3 a) { return 1.0f / fmaxf(sqrtf(f3dot(a, a)), EPSF); }

// max of the three corner cosines of triangle (a,b,c) — matches reference _max_cos
__device__ __forceinline__ float max_cos(F3 a, F3 b, F3 c) {
    F3 ab = f3sub(b, a), ac = f3sub(c, a), bc = f3sub(c, b);
    float rab = rlen(ab), rac = rlen(ac), rbc = rlen(bc);
    float c1 =  f3dot(ab, ac) * rab * rac;   // angle at a
    float c2 = -f3dot(ab, bc) * rab * rbc;   // angle at b
    float c3 =  f3dot(ac, bc) * rac * rbc;   // angle at c
    return fmaxf(c1, fmaxf(c2, c3));
}

__device__ __forceinline__ F3 ldv(const float* __restrict__ v, int i) {
    const float* p = v + (size_t)i * 3;
    return {p[0], p[1], p[2]};
}

// ---------------- K0: passthrough copy of verts ----------------
__global__ void k_copy_verts(const float* __restrict__ src, float* __restrict__ dst, int n) {
    int i = blockIdx.x * blockDim.x + threadIdx.x;
    int n4 = n >> 2;
    if (i < n4)
        reinterpret_cast<float4*>(dst)[i] = reinterpret_cast<const float4*>(src)[i];
    if (i == 0)
        for (int k = n4 * 4; k < n; ++k) dst[k] = src[k];
}

// ---------------- init: zero counters ----------------
__global__ void k_zero(uint32_t* __restrict__ p, int n) {
    int i = blockIdx.x * blockDim.x + threadIdx.x;
    if (i < n) p[i] = 0u;
}

// ---------------- K1: classify quads, per-block histogram ----------------
__global__ void k_classify(const float* __restrict__ verts,
                           const int* __restrict__ quads,
                           uint8_t* __restrict__ grp,
                           uint32_t* __restrict__ blockHist,
                           uint32_t* __restrict__ groupTotal,
                           int Q) {
    __shared__ uint32_t hist[4];
    const int tid = threadIdx.x;
    if (tid < 4) hist[tid] = 0u;
    __syncthreads();

    const int q = blockIdx.x * TB + tid;
    if (q < Q) {
        int4 qi = reinterpret_cast<const int4*>(quads)[q];   // coalesced b128
        F3 v0 = ldv(verts, qi.x), v1 = ldv(verts, qi.y);
        F3 v2 = ldv(verts, qi.z), v3 = ldv(verts, qi.w);

        F3 e10 = f3sub(v1, v0), e20 = f3sub(v2, v0), e30 = f3sub(v3, v0);
        F3 e21 = f3sub(v2, v1), e31 = f3sub(v3, v1);
        bool concave =
            (f3dot(f3cross(e10, e20), f3cross(e20, e30)) < 0.0f) ||
            (f3dot(f3cross(e10, e30), f3cross(e21, e31)) < 0.0f);

        float a1 = fmaxf(max_cos(v0, v1, v3), max_cos(v1, v2, v3));  // config 1
        float a2 = fmaxf(max_cos(v0, v1, v2), max_cos(v0, v2, v3));  // config 2
        bool ge = (a1 >= a2);

        int g = concave ? (ge ? 0 : 1) : (ge ? 3 : 2);
        grp[q] = (uint8_t)g;
        atomicAdd(&hist[g], 1u);
    }
    __syncthreads();
    if (tid < 4) {
        uint32_t c = hist[tid];
        blockHist[(size_t)blockIdx.x * 4 + tid] = c;
        if (c) atomicAdd(&groupTotal[tid], c);
    }
}

// ---------------- K2: per-group exclusive scan of per-block counts ----------------
__global__ void k_scan(uint32_t* __restrict__ blockHist,
                       const uint32_t* __restrict__ groupTotal,
                       int numBlocks) {
    const int g = blockIdx.x;        // one block per group (grid = 4)
    const int tid = threadIdx.x;
    uint32_t base = 0;
    for (int i = 0; i < g; ++i) base += groupTotal[i];

    __shared__ uint32_t tmp[TB];
    uint32_t carry = base;
    for (int start = 0; start < numBlocks; start += TB) {
        int i = start + tid;
        uint32_t v = (i < numBlocks) ? blockHist[(size_t)i * 4 + g] : 0u;
        tmp[tid] = v;
        __syncthreads();
        for (int off = 1; off < TB; off <<= 1) {
            uint32_t t = (tid >= off) ? tmp[tid - off] : 0u;
            __syncthreads();
            tmp[tid] += t;
            __syncthreads();
        }
        uint32_t incl = tmp[tid];
        uint32_t total = tmp[TB - 1];
        if (i < numBlocks) blockHist[(size_t)i * 4 + g] = carry + incl - v; // exclusive + base
        __syncthreads();
        carry += total;
    }
}

// ---------------- K3: stable-rank scatter of triangles ----------------
// Stages the block's quads into LDS via the CDNA5 async global->LDS path.
__global__ void k_scatter(const int* __restrict__ quads,
                          const uint8_t* __restrict__ grp,
                          const uint32_t* __restrict__ blockHist,
                          float* __restrict__ faces,
                          int Q) {
    __shared__ int qlds[TB * 4];
    __shared__ uint32_t waveCnt[TB / 32][4];

    const int tid = threadIdx.x;
    const int q = blockIdx.x * TB + tid;

    // Async stage: quads[blockBase .. blockBase+256) -> LDS (per-lane b128, ASYNCcnt)
    if (q < Q) {
        uint32_t ldsAddr = (uint32_t)(uintptr_t)(&qlds[tid * 4]);
        uint64_t gaddr = (uint64_t)(uintptr_t)(quads + (size_t)q * 4);
        asm volatile("global_load_async_to_lds_b128 %0, %1, off"
                     :: "v"(ldsAddr), "v"(gaddr)
                     : "memory");
    }
#if __has_builtin(__builtin_amdgcn_s_wait_asynccnt)
    __builtin_amdgcn_s_wait_asynccnt(0);
#else
    asm volatile("s_wait_asynccnt 0x0" ::: "memory");
#endif
    __syncthreads();

    const int lane = tid & 31;   // wave32
    const int wave = tid >> 5;   // 8 waves per block
    const int myg = (q < Q) ? (int)grp[q] : 4;

    // Uniform control flow for ballots
    uint64_t b0 = __ballot(myg == 0);
    uint64_t b1 = __ballot(myg == 1);
    uint64_t b2 = __ballot(myg == 2);
    uint64_t b3 = __ballot(myg == 3);
    if (lane == 0) {
        waveCnt[wave][0] = (uint32_t)__popcll(b0);
        waveCnt[wave][1] = (uint32_t)__popcll(b1);
        waveCnt[wave][2] = (uint32_t)__popcll(b2);
        waveCnt[wave][3] = (uint32_t)__popcll(b3);
    }
    __syncthreads();

    if (q < Q) {
        uint64_t mymask = (myg == 0) ? b0 : (myg == 1) ? b1 : (myg == 2) ? b2 : b3;
        uint32_t rank = (uint32_t)__popcll(mymask & ((1ull << lane) - 1ull));
        for (int w = 0; w < wave; ++w) rank += waveCnt[w][myg];
        uint32_t dest = blockHist[(size_t)blockIdx.x * 4 + myg] + rank; // stable position

        int a0 = qlds[tid * 4 + 0], a1 = qlds[tid * 4 + 1];
        int a2 = qlds[tid * 4 + 2], a3 = qlds[tid * 4 + 3];

        // use_cfg1 <=> group in {0,2};  IDX1 = (0,1,3),(1,2,3); IDX2 = (0,1,2),(0,2,3)
        bool cfg1 = (myg == 0) | (myg == 2);
        int t0, t1, t2, t3, t4, t5;
        if (cfg1) { t0 = a0; t1 = a1; t2 = a3; t3 = a1; t4 = a2; t5 = a3; }
        else      { t0 = a0; t1 = a1; t2 = a2; t3 = a0; t4 = a2; t5 = a3; }

        float* p = faces + (size_t)dest * 6;   // rows 2*dest, 2*dest+1
        p[0] = (float)t0; p[1] = (float)t1; p[2] = (float)t2;
        p[3] = (float)t3; p[4] = (float)t4; p[5] = (float)t5;
    }
}

// ---------------- host ----------------
extern "C" void kernel_launch(void* const* d_in, const int* in_sizes, int n_in,
                              void* d_out, int out_size, void* d_ws, size_t ws_size,
                              hipStream_t stream) {
    const float* verts = (const float*)d_in[0];
    const int* quads   = (const int*)d_in[1];
    const int V3 = in_sizes[0];         // 3 * V
    const int Q  = in_sizes[1] / 4;
    const int numBlocks = (Q + TB - 1) / TB;

    // workspace layout: [groupTotal(4 u32)][blockHist(numBlocks*4 u32)][grp(Q u8)]
    uint32_t* groupTotal = (uint32_t*)d_ws;
    uint32_t* blockHist  = groupTotal + 4;
    uint8_t*  grp        = (uint8_t*)(blockHist + (size_t)numBlocks * 4);

    float* outVerts = (float*)d_out;
    float* outFaces = outVerts + V3;

    int nz = 4 + numBlocks * 4;
    k_zero<<<(nz + TB - 1) / TB, TB, 0, stream>>>(groupTotal, nz);
    k_copy_verts<<<((V3 >> 2) + TB - 1) / TB, TB, 0, stream>>>(verts, outVerts, V3);
    k_classify<<<numBlocks, TB, 0, stream>>>(verts, quads, grp, blockHist, groupTotal, Q);
    k_scan<<<4, TB, 0, stream>>>(blockHist, groupTotal, numBlocks);
    k_scatter<<<numBlocks, TB, 0, stream>>>(quads, grp, blockHist, outFaces, Q);
}